// CRF_88072599371936
// MI455X (gfx1250) — compile-verified
//
#include <hip/hip_runtime.h>
#include <cstdint>

// Problem constants (match reference setup_inputs()).
#define BB 512
#define SS 512
#define TT 64
#define CH 16               // emission steps staged per chunk (4 KB)
#define NCH (SS / CH)       // 32 chunks
#define NLD ((CH * TT * 4) / (32 * 16))   // async b128 ops per chunk = 8

// ---- gfx1250 async global->LDS path (ASYNCcnt), with safe fallback --------
#if __has_builtin(__builtin_amdgcn_global_load_async_to_lds_b128) && \
    __has_builtin(__builtin_amdgcn_s_wait_asynccnt)
#define HAVE_ASYNC 1
#endif

typedef int v4i __attribute__((ext_vector_type(4)));
typedef __attribute__((address_space(1))) v4i gv4i;   // global v4i
typedef __attribute__((address_space(3))) v4i sv4i;   // LDS v4i

#ifdef HAVE_ASYNC
#define AWAIT(n) __builtin_amdgcn_s_wait_asynccnt(n)
#else
#define AWAIT(n)
#endif

__device__ __forceinline__ void stage_chunk(const float* __restrict__ gbase,
                                            float* __restrict__ lbuf, int lane) {
#pragma unroll
  for (int k = 0; k < NLD; ++k) {
    const float* g = gbase + (k * 32 + lane) * 4;
    float* d = lbuf + (k * 32 + lane) * 4;
#ifdef HAVE_ASYNC
    __builtin_amdgcn_global_load_async_to_lds_b128(
        (gv4i*)(uintptr_t)g,
        (sv4i*)(uint32_t)(uintptr_t)d, /*offset=*/0, /*cpol=*/0);
#else
    float4 v = *(const float4*)g;
    *(float4*)d = v;
#endif
  }
}

__device__ __forceinline__ float bcast_f32(float v, int srclane) {
  return __int_as_float(__builtin_amdgcn_readlane(__float_as_int(v), srclane));
}

// One wave32 per batch. Lane l owns tags j0=2l, j1=2l+1.
__global__ void __launch_bounds__(32)
viterbi_kernel(const float* __restrict__ em,   // [B,S,T]
               const float* __restrict__ tr,   // [T,T]
               int* __restrict__ out) {        // [B,S] int32 tags
  __shared__ unsigned short hist16[(SS - 1) * 32];   // backptr bytes, packed 2/lane
  __shared__ float ebuf[2][CH * TT];                 // double-buffered emissions
  __shared__ int tags_lds[SS];

  const int l = threadIdx.x;
  const int b = blockIdx.x;
  const float* embase = em + (size_t)b * SS * TT;

  // Transition columns j0=2l, j1=2l+1 held entirely in VGPRs (128 regs/lane).
  float tA[TT], tB[TT];
#pragma unroll
  for (int i = 0; i < TT; ++i) {
    float2 tv = *(const float2*)&tr[i * TT + 2 * l];
    tA[i] = tv.x;
    tB[i] = tv.y;
  }

  // Kick off chunk 0 staging.
  stage_chunk(embase, ebuf[0], l);

  float s0 = 0.0f, s1 = 0.0f;   // viterbi scores for tags 2l, 2l+1

#pragma unroll 1
  for (int c = 0; c < NCH; ++c) {
    if (c + 1 < NCH) {
      stage_chunk(embase + (size_t)(c + 1) * CH * TT, ebuf[(c + 1) & 1], l);
      AWAIT(NLD);   // chunk c's older loads complete; next chunk's 8 may fly
    } else {
      AWAIT(0);
    }
    asm volatile("" ::: "memory");  // keep ds reads after the wait

    const float* eb = ebuf[c & 1];
#pragma unroll 1
    for (int sc = 0; sc < CH; ++sc) {
      const int s = c * CH + sc;
      float2 e = *(const float2*)&eb[sc * TT + 2 * l];
      if (s == 0) { s0 = e.x; s1 = e.y; continue; }

      float best0 = -INFINITY, best1 = -INFINITY;
      int a0 = 0, a1 = 0;
#pragma unroll
      for (int i = 0; i < TT; ++i) {
        // score[i] broadcast: constant-lane v_readlane -> SGPR operand adds
        float si = bcast_f32((i & 1) ? s1 : s0, i >> 1);
        float c0 = si + tA[i];
        float c1 = si + tB[i];
        if (c0 > best0) { best0 = c0; a0 = i; }   // strict > => first-occurrence argmax
        if (c1 > best1) { best1 = c1; a1 = i; }
      }
      s0 = best0 + e.x;
      s1 = best1 + e.y;
      hist16[(s - 1) * 32 + l] = (unsigned short)(a0 | (a1 << 8));
    }
  }

  // ---- argmax over final scores (lowest index wins ties, like jnp.argmax) --
  float bv;
  int bi;
  if (s0 >= s1) { bv = s0; bi = 2 * l; } else { bv = s1; bi = 2 * l + 1; }
#pragma unroll
  for (int off = 16; off >= 1; off >>= 1) {
    float ov = __shfl_xor(bv, off, 32);
    int oi = __shfl_xor(bi, off, 32);
    if (ov > bv || (ov == bv && oi < bi)) { bv = ov; bi = oi; }
  }
  int best = bi;  // uniform across the wave
  tags_lds[SS - 1] = best;

  // ---- backtrace entirely from LDS (ds_load row + v_readlane select) -------
#pragma unroll 1
  for (int t = SS - 2; t >= 0; --t) {
    unsigned v = hist16[t * 32 + l];                       // lane holds bytes 2l,2l+1
    unsigned sel = (unsigned)__builtin_amdgcn_readlane((int)v, best >> 1);
    best = (best & 1) ? (int)((sel >> 8) & 0xff) : (int)(sel & 0xff);
    tags_lds[t] = best;   // all lanes write the same value
  }
  __syncthreads();

  // Coalesced int32 output for this batch.
  int* outb = out + (size_t)b * SS;
#pragma unroll
  for (int k = 0; k < SS / 32; ++k) outb[k * 32 + l] = tags_lds[k * 32 + l];
}

extern "C" void kernel_launch(void* const* d_in, const int* in_sizes, int n_in,
                              void* d_out, int out_size, void* d_ws, size_t ws_size,
                              hipStream_t stream) {
  const float* emissions = (const float*)d_in[0];     // [512,512,64] f32
  const float* transitions = (const float*)d_in[1];   // [64,64] f32
  int* out = (int*)d_out;                             // [512,512] int32 tags
  (void)in_sizes; (void)n_in; (void)out_size; (void)d_ws; (void)ws_size;

  viterbi_kernel<<<dim3(BB), dim3(32), 0, stream>>>(emissions, transitions, out);
}